// Net_48524540510802
// MI455X (gfx1250) — compile-verified
//
#include <hip/hip_runtime.h>

// GAT fused kernel for MI455X (gfx1250, wave32, WMMA).
// Dominant cost: [N*K,128]x[128,256] GEMM (42 GFLOP) fused with attention so
// the 655MB h_j intermediate never leaves the WGP. HBM floor ~340MB @23.3TB/s.

#define KNBR 32
#define FDIM 128
#define NH   4
#define DH   64
#define HD   256
#define NCLS 40

#define NBR_STRIDE 132   // padded LDS row stride (floats) for neighbor tile
#define HJ_STRIDE  260   // padded LDS row stride (floats) for h_j tile

typedef float v2f __attribute__((ext_vector_type(2)));
typedef float v4f __attribute__((ext_vector_type(4)));
typedef float v8f __attribute__((ext_vector_type(8)));

__global__ __launch_bounds__(256) void gat_fused(
    const float* __restrict__ nodef_g,   // [N,128]
    const float* __restrict__ nbr_g,     // [N,32,128]
    const float* __restrict__ Wg,        // [128,256]
    const float* __restrict__ a_src,     // [4,64]
    const float* __restrict__ a_dst,     // [4,64]
    const float* __restrict__ cls_w,     // [256,40]
    float* __restrict__ out,             // [N,40]
    int n_nodes)
{
  __shared__ float uSh[KNBR * HJ_STRIDE]; // neighbor tile (stride 132) then h_j (stride 260)
  __shared__ float nodefS[FDIM];
  __shared__ float qS[FDIM * NH];         // q[f][h] = sum_d W[f][64h+d]*a_src[h][d]
  __shared__ float adstS[HD];
  __shared__ float eiS[NH];
  __shared__ float alphaS[NH * KNBR];
  __shared__ float outvS[HD];
  __shared__ float redS[4 * 64];

  const int t    = threadIdx.x;
  const int lane = t & 31;
  const int w    = t >> 5;      // wave id 0..7
  const int l16  = lane & 15;
  const int lhi  = lane >> 4;   // 0 or 1 (half-wave)

  // ---------------- per-workgroup prologue (amortized over ~10 nodes) -------
  if (t < HD) adstS[t] = a_dst[t];

  { // q[f][h]: fold a_src through W so h_i is never materialized
    int f  = t & 127;
    int h0 = t >> 7;                      // 0 or 1
    #pragma unroll
    for (int hh = 0; hh < 2; ++hh) {
      int h = h0 + 2 * hh;
      const float* wr = Wg + f * HD + h * DH;
      const float* ar = a_src + h * DH;
      float s = 0.f;
      #pragma unroll 8
      for (int d = 0; d < DH; ++d) s += wr[d] * ar[d];
      qS[f * NH + h] = s;
    }
  }

  // B fragments of W for this wave's 32 output columns: resident in VGPRs.
  // V_WMMA_F32_16X16X4_F32 B(4x16) layout: v0 = {K0 | K2}, v1 = {K1 | K3}
  // (lanes 0-15 | 16-31), column = lane&15.
  const int cbase = w * 32;
  v2f Bf[32][2];
  #pragma unroll
  for (int kt = 0; kt < 32; ++kt) {
    #pragma unroll
    for (int nt = 0; nt < 2; ++nt) {
      int col = cbase + nt * 16 + l16;
      int rb  = kt * 4 + lhi * 2;
      v2f b;
      b.x = Wg[rb * HD + col];
      b.y = Wg[(rb + 1) * HD + col];
      Bf[kt][nt] = b;
    }
  }
  __syncthreads();

  // ---------------- node loop ----------------------------------------------
  for (int node = blockIdx.x; node < n_nodes; node += gridDim.x) {
    // Stage neighbor tile [32][128] coalesced (float4) into padded LDS.
    const float* nb = nbr_g + (size_t)node * (KNBR * FDIM);
    #pragma unroll
    for (int i = 0; i < 4; ++i) {
      int fi = i * 256 + t;     // float4 index 0..1023
      int fl = fi * 4;
      int k  = fl >> 7;
      int f  = fl & 127;
      v4f v = *(const v4f*)(nb + fl);
      *(v4f*)(&uSh[k * NBR_STRIDE + f]) = v;
    }
    if (t < FDIM) nodefS[t] = nodef_g[(size_t)node * FDIM + t];
    __syncthreads();

    { // prefetch next node's neighbor tile (64B per thread covers 16KB)
      int nn = node + gridDim.x;
      if (nn < n_nodes)
        __builtin_prefetch((const char*)(nbr_g + (size_t)nn * (KNBR * FDIM)) + t * 64, 0, 3);
    }

    // GEMM: h_j(32x256) = nbr(32x128) @ W(128x256); this wave: 32 columns.
    // A(16x4) layout: lane&15 = row, v0/v1 = {K0,K1} (lanes 0-15), {K2,K3} (16-31).
    v8f acc00 = {}, acc01 = {}, acc10 = {}, acc11 = {};
    #pragma unroll
    for (int kt = 0; kt < 32; ++kt) {
      int dw = kt * 4 + lhi * 2;
      v2f a0 = *(const v2f*)(&uSh[(l16)      * NBR_STRIDE + dw]);
      v2f a1 = *(const v2f*)(&uSh[(16 + l16) * NBR_STRIDE + dw]);
      acc00 = __builtin_amdgcn_wmma_f32_16x16x4_f32(false, a0, false, Bf[kt][0], (short)0, acc00, false, false);
      acc01 = __builtin_amdgcn_wmma_f32_16x16x4_f32(false, a0, false, Bf[kt][1], (short)0, acc01, false, false);
      acc10 = __builtin_amdgcn_wmma_f32_16x16x4_f32(false, a1, false, Bf[kt][0], (short)0, acc10, false, false);
      acc11 = __builtin_amdgcn_wmma_f32_16x16x4_f32(false, a1, false, Bf[kt][1], (short)0, acc11, false, false);
    }
    __syncthreads();  // all waves done reading nbr tile; reuse region as h_j

    { // scatter C fragments to h_j LDS (stride 260 -> conflict-free halves)
      int colA  = cbase + l16;
      int colB  = cbase + 16 + l16;
      int rbase = lhi * 8;
      #pragma unroll
      for (int r = 0; r < 8; ++r) {
        uSh[(rbase + r) * HJ_STRIDE + colA]      = acc00[r];
        uSh[(rbase + r) * HJ_STRIDE + colB]      = acc01[r];
        uSh[(16 + rbase + r) * HJ_STRIDE + colA] = acc10[r];
        uSh[(16 + rbase + r) * HJ_STRIDE + colB] = acc11[r];
      }
    }

    // e_i[h] = nodef . q[:,h]  (waves 4..7, one head each, wave reduction)
    if (w >= 4) {
      int h = w - 4;
      float s = 0.f;
      #pragma unroll
      for (int j = 0; j < 4; ++j) {
        int f = lane + 32 * j;
        s += nodefS[f] * qS[f * NH + h];
      }
      #pragma unroll
      for (int off = 16; off > 0; off >>= 1) s += __shfl_xor(s, off, 32);
      if (lane == 0) eiS[h] = s;
    }
    __syncthreads();

    // e_j + LeakyReLU + softmax over K (waves 0..3: wave = head, lane = k)
    if (w < 4) {
      int h = w, k = lane;
      const float* hjrow = &uSh[k * HJ_STRIDE + h * DH];
      const float* ad    = &adstS[h * DH];
      float e = 0.f;
      #pragma unroll 8
      for (int d = 0; d < DH; ++d) e += hjrow[d] * ad[d];
      e += eiS[h];
      e = (e > 0.f) ? e : 0.2f * e;       // leaky_relu, slope 0.2
      float m = e;
      #pragma unroll
      for (int off = 16; off > 0; off >>= 1) m = fmaxf(m, __shfl_xor(m, off, 32));
      float ex = __expf(e - m);
      float s = ex;
      #pragma unroll
      for (int off = 16; off > 0; off >>= 1) s += __shfl_xor(s, off, 32);
      alphaS[h * KNBR + k] = ex / s;
    }
    __syncthreads();

    // aggregation: outv[n] = sum_k alpha[k, h(n)] * h_j[k, n]
    {
      int n = t;
      int h = n >> 6;
      float s = 0.f;
      #pragma unroll 8
      for (int k = 0; k < KNBR; ++k)
        s += alphaS[h * KNBR + k] * uSh[k * HJ_STRIDE + n];
      outvS[n] = s;
    }
    __syncthreads();

    // classifier: out[c] = sum_j outv[j] * weight[j][c], split 4 ways over j
    {
      int part = t >> 6;
      int c    = t & 63;
      if (c < NCLS) {
        float s = 0.f;
        #pragma unroll 8
        for (int j = 0; j < 64; ++j) {
          int jj = part * 64 + j;
          s += outvS[jj] * cls_w[jj * NCLS + c];
        }
        redS[part * 64 + c] = s;
      }
    }
    __syncthreads();
    if (t < NCLS) {
      float s = redS[t] + redS[64 + t] + redS[128 + t] + redS[192 + t];
      out[(size_t)node * NCLS + t] = s;
    }
    __syncthreads();
  }
}

extern "C" void kernel_launch(void* const* d_in, const int* in_sizes, int n_in,
                              void* d_out, int out_size, void* d_ws, size_t ws_size,
                              hipStream_t stream) {
  const float* nodef = (const float*)d_in[0];
  const float* nbr   = (const float*)d_in[1];
  const float* Wg    = (const float*)d_in[2];
  const float* asrc  = (const float*)d_in[3];
  const float* adst  = (const float*)d_in[4];
  const float* cw    = (const float*)d_in[5];
  float* outp = (float*)d_out;

  int n = in_sizes[0] / FDIM;   // N nodes
  int blocks = 2048;            // persistent-ish: amortize W-in-VGPR preload
  if (blocks > n) blocks = n;
  gat_fused<<<blocks, 256, 0, stream>>>(nodef, nbr, Wg, asrc, adst, cw, outp, n);
}